// PrimalPerturbationBlock_5016521802239
// MI455X (gfx1250) — compile-verified
//
#include <hip/hip_runtime.h>
#include <math.h>

typedef float v2f __attribute__((ext_vector_type(2)));
typedef float v4f __attribute__((ext_vector_type(4)));
typedef float v8f __attribute__((ext_vector_type(8)));

// ---------------------------------------------------------------------------
// WMMA GEMM: C[M x 16] = A[M x KPAD] @ W[K x 16] (+ bias) (+ ReLU), fp32 via
// V_WMMA_F32_16X16X4_F32. One wave = one 16x16 output tile. KPAD is the
// compile-time padded row stride of A (multiple of 4, pad columns zeroed by
// the builder kernels). W rows >= K are read as zero (preloaded, hoisted).
// ---------------------------------------------------------------------------
template <int KPAD>
__global__ void gemm16_wmma(const float* __restrict__ A, int K, int M,
                            const float* __restrict__ W,
                            const float* __restrict__ bias,
                            float* __restrict__ C, int relu)
{
    constexpr int KS = KPAD / 4;
    const int wave = threadIdx.x >> 5;
    const int lane = threadIdx.x & 31;
    const int tile = blockIdx.x * (blockDim.x >> 5) + wave;
    if (tile * 16 >= M) return;                 // wave-uniform exit: EXEC stays full
    const int half = lane >> 4;                 // 0: lanes 0-15, 1: lanes 16-31
    const int rc   = lane & 15;                 // A row within tile / B,C column
    const long row = (long)tile * 16 + rc;

    // Preload B fragments (hoisted; only place with K guards).
    v2f bf[KS];
#pragma unroll
    for (int s = 0; s < KS; ++s) {
        const int ka = 4 * s + 2 * half;
        bf[s].x = (ka     < K) ? W[ka * 16 + rc]       : 0.f;
        bf[s].y = (ka + 1 < K) ? W[(ka + 1) * 16 + rc] : 0.f;
    }

    const float* Ap = A + row * KPAD + 2 * half;
    v8f acc = {0.f, 0.f, 0.f, 0.f, 0.f, 0.f, 0.f, 0.f};
#pragma unroll
    for (int s = 0; s < KS; ++s) {
        const v2f a = *(const v2f*)(Ap + 4 * s);     // global_load_b64, unconditional
        acc = __builtin_amdgcn_wmma_f32_16x16x4_f32(
            false, a, false, bf[s], (short)0, acc, false, false);
    }

    const float bb = bias ? bias[rc] : 0.f;
#pragma unroll
    for (int r = 0; r < 8; ++r) {
        const long m = (long)tile * 16 + r + 8 * half;
        float v = acc[r] + bb;
        if (relu) v = fmaxf(v, 0.f);
        C[m * 16 + rc] = v;
    }
}

// ---------------------------------------------------------------------------
// Elementwise builders (write padded rows; pad columns zeroed)
// ---------------------------------------------------------------------------
__global__ void build_node_comb(const float* __restrict__ learned,
                                const float* __restrict__ lp, int lpdim,
                                long total, int wpad, float* __restrict__ out)
{
    long i = (long)blockIdx.x * blockDim.x + threadIdx.x;
    if (i >= total) return;
    long node = i / wpad; int c = (int)(i % wpad);
    float v;
    if (c < 16)            v = learned[node * 16 + c];
    else if (c < 16 + lpdim) v = lp[node * lpdim + (c - 16)];
    else                   v = 0.f;
    out[i] = v;
}

__global__ void build_edge_comb(const float* __restrict__ learned,
                                const float* __restrict__ lo,
                                const float* __restrict__ hi,
                                const float* __restrict__ lp, int lpdim,
                                long total, int wpad, float* __restrict__ out)
{
    long i = (long)blockIdx.x * blockDim.x + threadIdx.x;
    if (i >= total) return;
    long e = i / wpad; int c = (int)(i % wpad);
    float v;
    if (c < 16)              v = learned[e * 16 + c];
    else if (c == 16)        v = lo[e];
    else if (c == 17)        v = hi[e];
    else if (c < 18 + lpdim) v = lp[e * lpdim + (c - 18)];
    else                     v = 0.f;
    out[i] = v;
}

__global__ void build_ein(const float* __restrict__ ec, int ecStride, int we,
                          const float* __restrict__ vsrc,
                          const float* __restrict__ cdst,
                          const int* __restrict__ sidx,
                          const int* __restrict__ didx,
                          long total, int wpad, float* __restrict__ out)
{
    long i = (long)blockIdx.x * blockDim.x + threadIdx.x;
    if (i >= total) return;
    long e = i / wpad; int c = (int)(i % wpad);
    float v;
    if (c < we)             v = ec[e * ecStride + c];
    else if (c < we + 16)   v = vsrc[(long)sidx[e] * 16 + (c - we)];
    else if (c < we + 32)   v = cdst[(long)didx[e] * 16 + (c - we - 16)];
    else                    v = 0.f;
    out[i] = v;
}

__global__ void add_deg(const int* __restrict__ idx, long n, float* __restrict__ deg)
{
    long i = (long)blockIdx.x * blockDim.x + threadIdx.x;
    if (i < n) atomicAdd(&deg[idx[i]], 1.0f);
}

// ---------------------------------------------------------------------------
// Attention (edge softmax per destination segment)
// ---------------------------------------------------------------------------
__device__ __forceinline__ unsigned f32_ordkey(float f) {
    unsigned b = __float_as_uint(f);
    return (b & 0x80000000u) ? ~b : (b | 0x80000000u);
}
__device__ __forceinline__ float ordkey_f32(unsigned k) {
    return (k & 0x80000000u) ? __uint_as_float(k ^ 0x80000000u) : __uint_as_float(~k);
}

__global__ void attn_logits(const float* __restrict__ q, const float* __restrict__ k,
                            const float* __restrict__ e,
                            const int* __restrict__ sidx, const int* __restrict__ didx,
                            long nE, float* __restrict__ logits, unsigned* __restrict__ smax)
{
    long i = (long)blockIdx.x * blockDim.x + threadIdx.x;
    if (i >= nE) return;
    long s = sidx[i], d = didx[i];
    const v4f* q4 = (const v4f*)(q + d * 16);
    const v4f* k4 = (const v4f*)(k + s * 16);
    const v4f* e4 = (const v4f*)(e + i * 16);
    float acc = 0.f;
#pragma unroll
    for (int j = 0; j < 4; ++j) {
        v4f qq = q4[j], kk = k4[j], ee = e4[j];
        acc += qq.x * (kk.x + ee.x) + qq.y * (kk.y + ee.y)
             + qq.z * (kk.z + ee.z) + qq.w * (kk.w + ee.w);
    }
    acc *= 0.25f;                                   // 1/sqrt(16)
    logits[i] = acc;
    atomicMax(&smax[d], f32_ordkey(acc));
}

__global__ void attn_exp(float* __restrict__ logits, const int* __restrict__ didx,
                         long nE, const unsigned* __restrict__ smax,
                         float* __restrict__ ssum)
{
    long i = (long)blockIdx.x * blockDim.x + threadIdx.x;
    if (i >= nE) return;
    long d = didx[i];
    float ex = __expf(logits[i] - ordkey_f32(smax[d]));
    logits[i] = ex;
    atomicAdd(&ssum[d], ex);
}

__global__ void attn_agg(const float* __restrict__ ex, const float* __restrict__ v,
                         const float* __restrict__ e,
                         const int* __restrict__ sidx, const int* __restrict__ didx,
                         long nE, const float* __restrict__ ssum, float* __restrict__ agg)
{
    long i = (long)blockIdx.x * blockDim.x + threadIdx.x;
    if (i >= nE) return;
    long s = sidx[i], d = didx[i];
    float alpha = ex[i] / (ssum[d] + 1e-16f);
    const v4f* v4 = (const v4f*)(v + s * 16);
    const v4f* e4 = (const v4f*)(e + i * 16);
    float* a = agg + d * 16;
#pragma unroll
    for (int j = 0; j < 4; ++j) {
        v4f vv = v4[j], ee = e4[j];
        atomicAdd(a + 4 * j + 0, alpha * (vv.x + ee.x));
        atomicAdd(a + 4 * j + 1, alpha * (vv.y + ee.y));
        atomicAdd(a + 4 * j + 2, alpha * (vv.z + ee.z));
        atomicAdd(a + 4 * j + 3, alpha * (vv.w + ee.w));
    }
}

__global__ void attn_fin(const float* __restrict__ agg, const float* __restrict__ deg,
                         const float* __restrict__ skip, long total,
                         float* __restrict__ outF)
{
    long i = (long)blockIdx.x * blockDim.x + threadIdx.x;
    if (i >= total) return;
    long node = i >> 4;
    outF[i] = fmaxf(agg[i] / fmaxf(deg[node], 1.f) + skip[i], 0.f);
}

// ---------------------------------------------------------------------------
// Perturbation head: pert = 0.1 * mlp2([edge_f(16) ++ edge_lp(4)]), 20->20->1
// ---------------------------------------------------------------------------
__global__ void pert_head(const float* __restrict__ ef, const float* __restrict__ elp,
                          const float* __restrict__ lo, const float* __restrict__ hi,
                          const float* __restrict__ w0, const float* __restrict__ b0,
                          const float* __restrict__ w1, const float* __restrict__ b1,
                          long nE, float* __restrict__ out)
{
    long i = (long)blockIdx.x * blockDim.x + threadIdx.x;
    if (i >= nE) return;
    float h[20];
#pragma unroll
    for (int j = 0; j < 20; ++j) h[j] = b0[j];
#pragma unroll
    for (int c = 0; c < 16; ++c) {
        float x = ef[i * 16 + c];
#pragma unroll
        for (int j = 0; j < 20; ++j) h[j] += x * w0[c * 20 + j];
    }
#pragma unroll
    for (int c = 0; c < 4; ++c) {
        float x = elp[i * 4 + c];
#pragma unroll
        for (int j = 0; j < 20; ++j) h[j] += x * w0[(16 + c) * 20 + j];
    }
    float p = b1[0];
#pragma unroll
    for (int j = 0; j < 20; ++j) p += fmaxf(h[j], 0.f) * w1[j];
    p *= 0.1f;
    out[2 * i]     = lo[i] + fmaxf( p + 5e-4f, 0.f);
    out[2 * i + 1] = hi[i] + fmaxf(-p + 5e-4f, 0.f);
}

// ---------------------------------------------------------------------------
// Host orchestration
// ---------------------------------------------------------------------------
struct TConvP {
    const float *ew, *kb, *kw, *qb, *qw, *sb, *sw, *vb, *vw;
};

extern "C" void kernel_launch(void* const* d_in, const int* in_sizes, int n_in,
                              void* d_out, int out_size, void* d_ws, size_t ws_size,
                              hipStream_t stream)
{
    if (n_in < 73) return;
    const long NV  = in_sizes[3] / 16;   // 100000
    const long NC  = in_sizes[4] / 16;   // 50000
    const long NE  = in_sizes[6];        // 1000000
    const int vlp  = (int)(in_sizes[0] / NV);    // 3
    const int clp  = (int)(in_sizes[1] / NC);    // 3
    const int elp  = (int)(in_sizes[2] / NE);    // 4
    const int WV   = 16 + vlp;           // 19
    const int WC   = 16 + clp;           // 19
    const int WE   = 16 + 2 + elp;       // 22
    const int WEIN = WE + 32;            // 54
    const int WVP = 20, WCP = 20, WEP = 24, WEINP = 56;   // padded strides

    const float* var_lp  = (const float*)d_in[0];
    const float* con_lp  = (const float*)d_in[1];
    const float* edge_lp = (const float*)d_in[2];
    const float* lo      = (const float*)d_in[6];
    const float* hi      = (const float*)d_in[7];
    const int*   srcIdx  = (const int*)d_in[8];
    const int*   dstIdx  = srcIdx + NE;

    auto F = [&](int i) { return (const float*)d_in[i]; };

    // ----- workspace bump allocator (256B aligned)
    char* base = (char*)d_ws; size_t off = 0;
    auto alloc = [&](size_t bytes) -> float* {
        float* p = (float*)(base + off);
        off = (off + bytes + 255) & ~(size_t)255;
        return p;
    };
    float* VARF  = alloc(NV * 16 * 4);
    float* CONF  = alloc(NC * 16 * 4);
    float* EDGEF = alloc(NE * 16 * 4);
    float* VARC  = alloc(NV * (size_t)WVP * 4);
    float* CONC  = alloc(NC * (size_t)WCP * 4);
    float* EDGEC = alloc(NE * (size_t)WEP * 4);
    float* BUFE  = alloc(NE * 16 * 4);   // edge proj e / edge-MLP hidden
    float* LOGIT = alloc(NE * 4);
    float* BUFQ  = alloc(NV * 16 * 4);
    float* BUFK  = alloc(NV * 16 * 4);
    float* BUFV  = alloc(NV * 16 * 4);
    float* BUFS  = alloc(NV * 16 * 4);
    unsigned* SMAX = (unsigned*)alloc(NV * 4);
    float* SSUM  = alloc(NV * 4);
    float* AGG   = alloc(NV * 16 * 4);
    float* DEGV  = alloc(NV * 4);
    float* DEGC  = alloc(NC * 4);
    float* EIN   = alloc(NE * (size_t)WEINP * 4);
    (void)ws_size;

    auto eb = [&](long n) { return dim3((unsigned)((n + 255) / 256)); };

    auto gemm = [&](const float* A, int kpad, int K, long M, const float* W,
                    const float* b, float* C, int relu) {
        long waves = (M + 15) / 16;
        dim3 grid((unsigned)((waves + 7) / 8));
        switch (kpad) {
        case 16: gemm16_wmma<16><<<grid, 256, 0, stream>>>(A, K, (int)M, W, b, C, relu); break;
        case 20: gemm16_wmma<20><<<grid, 256, 0, stream>>>(A, K, (int)M, W, b, C, relu); break;
        case 24: gemm16_wmma<24><<<grid, 256, 0, stream>>>(A, K, (int)M, W, b, C, relu); break;
        default: gemm16_wmma<56><<<grid, 256, 0, stream>>>(A, K, (int)M, W, b, C, relu); break;
        }
    };

    // initial learned features
    hipMemcpyAsync(VARF,  d_in[3], NV * 16 * 4, hipMemcpyDeviceToDevice, stream);
    hipMemcpyAsync(CONF,  d_in[4], NC * 16 * 4, hipMemcpyDeviceToDevice, stream);
    hipMemcpyAsync(EDGEF, d_in[5], NE * 16 * 4, hipMemcpyDeviceToDevice, stream);

    // degrees (constant across layers)
    hipMemsetAsync(DEGV, 0, NV * 4, stream);
    hipMemsetAsync(DEGC, 0, NC * 4, stream);
    add_deg<<<eb(NE), 256, 0, stream>>>(srcIdx, NE, DEGV);
    add_deg<<<eb(NE), 256, 0, stream>>>(dstIdx, NE, DEGC);

    auto attn = [&](const TConvP& tp,
                    const float* srcComb, int wSrcP, int wSrc, long nSrc,
                    const float* dstComb, int wDstP, int wDst, long nDst,
                    const int* sI, const int* dI, const float* deg, float* outF) {
        gemm(srcComb, wSrcP, wSrc, nSrc, tp.kw, tp.kb, BUFK, 0);
        gemm(srcComb, wSrcP, wSrc, nSrc, tp.vw, tp.vb, BUFV, 0);
        gemm(dstComb, wDstP, wDst, nDst, tp.qw, tp.qb, BUFQ, 0);
        gemm(dstComb, wDstP, wDst, nDst, tp.sw, tp.sb, BUFS, 0);
        gemm(EDGEC,   WEP,   WE,   NE,   tp.ew, nullptr, BUFE, 0);
        hipMemsetAsync(SMAX, 0, nDst * 4, stream);
        hipMemsetAsync(SSUM, 0, nDst * 4, stream);
        hipMemsetAsync(AGG,  0, nDst * 16 * 4, stream);
        attn_logits<<<eb(NE), 256, 0, stream>>>(BUFQ, BUFK, BUFE, sI, dI, NE, LOGIT, SMAX);
        attn_exp<<<eb(NE), 256, 0, stream>>>(LOGIT, dI, NE, SMAX, SSUM);
        attn_agg<<<eb(NE), 256, 0, stream>>>(LOGIT, BUFV, BUFE, sI, dI, NE, SSUM, AGG);
        attn_fin<<<eb(nDst * 16), 256, 0, stream>>>(AGG, deg, BUFS, nDst * 16, outF);
    };

    for (int L = 0; L < 2; ++L) {
        const int B = 9 + L * 30;
        // pytree order (dict keys sorted): con_upd{e.w,k.b,k.w,q.b,q.w,skip.b,skip.w,v.b,v.w},
        // edge_upd{con_c[b,w,b,w], edge[b,w,b,w], var_c[b,w,b,w]}, var_upd{...}
        TConvP conU = { F(B+0), F(B+1), F(B+2), F(B+3), F(B+4), F(B+5), F(B+6), F(B+7), F(B+8) };
        const float *cc0b = F(B+9),  *cc0w = F(B+10), *cc1b = F(B+11), *cc1w = F(B+12);
        const float *ee0b = F(B+13), *ee0w = F(B+14), *ee1b = F(B+15), *ee1w = F(B+16);
        const float *vc0b = F(B+17), *vc0w = F(B+18), *vc1b = F(B+19), *vc1w = F(B+20);
        TConvP varU = { F(B+21), F(B+22), F(B+23), F(B+24), F(B+25), F(B+26), F(B+27), F(B+28), F(B+29) };

        build_node_comb<<<eb(NV * WVP), 256, 0, stream>>>(VARF, var_lp, vlp, NV * WVP, WVP, VARC);
        build_node_comb<<<eb(NC * WCP), 256, 0, stream>>>(CONF, con_lp, clp, NC * WCP, WCP, CONC);
        build_edge_comb<<<eb(NE * WEP), 256, 0, stream>>>(EDGEF, lo, hi, edge_lp, elp, NE * WEP, WEP, EDGEC);

        // con update: var -> con
        attn(conU, VARC, WVP, WV, NV, CONC, WCP, WC, NC, srcIdx, dstIdx, DEGC, CONF);
        build_node_comb<<<eb(NC * WCP), 256, 0, stream>>>(CONF, con_lp, clp, NC * WCP, WCP, CONC);

        // var update: con -> var (src features = new con, dst features = old var)
        attn(varU, CONC, WCP, WC, NC, VARC, WVP, WV, NV, dstIdx, srcIdx, DEGV, VARF);
        build_node_comb<<<eb(NV * WVP), 256, 0, stream>>>(VARF, var_lp, vlp, NV * WVP, WVP, VARC);

        // edge update
        gemm(VARC, WVP, WV, NV, vc0w, vc0b, BUFK, 1);        // var_c hidden (relu)
        gemm(BUFK, 16,  16, NV, vc1w, vc1b, BUFV, 1);        // var_c out (relu)
        gemm(CONC, WCP, WC, NC, cc0w, cc0b, BUFQ, 1);        // con_c hidden
        gemm(BUFQ, 16,  16, NC, cc1w, cc1b, BUFS, 1);        // con_c out
        build_ein<<<eb(NE * (long)WEINP), 256, 0, stream>>>(
            EDGEC, WEP, WE, BUFV, BUFS, srcIdx, dstIdx, NE * (long)WEINP, WEINP, EIN);
        gemm(EIN,  WEINP, WEIN, NE, ee0w, ee0b, BUFE, 1);    // edge MLP hidden (relu)
        gemm(BUFE, 16,    16,   NE, ee1w, ee1b, EDGEF, 1);   // edge MLP out (+outer relu)
    }

    // perturbation head -> d_out [NE, 2]
    const float* pb0 = F(69); const float* pw0 = F(70);
    const float* pb1 = F(71); const float* pw1 = F(72);
    pert_head<<<eb(NE), 256, 0, stream>>>(EDGEF, edge_lp, lo, hi, pw0, pb0, pw1, pb1,
                                          NE, (float*)d_out);
    (void)out_size;
}